// Aggregator_309237645952
// MI455X (gfx1250) — compile-verified
//
#include <hip/hip_runtime.h>
#include <hip/hip_bf16.h>

// ---------------------------------------------------------------------------
// MI455X (gfx1250) hyperbolic edge-aggregation.
//   - wave32: one wave per edge, lane owns 2 of d=64 dims (float2 / b64 loads)
//   - dot products: 5-step shfl_xor butterfly (log2(32))
//   - relation table (32x64 f32 = 8KB) staged to LDS via the CDNA5 Tensor
//     Data Mover (tensor_load_to_lds + s_wait_tensorcnt)
//   - scatter-mean via global f32 atomics into d_out, counts in d_ws
// Memory-bound: ~0.62 GB traffic -> ~27us floor at 23.3 TB/s. No GEMM
// structure, so WMMA is inapplicable; TDM/async is the gfx1250 path here.
// ---------------------------------------------------------------------------

typedef unsigned int u32;
typedef unsigned long long u64;
typedef u32 u32x4 __attribute__((ext_vector_type(4)));
typedef int  i32x4 __attribute__((ext_vector_type(4)));
typedef int  i32x8 __attribute__((ext_vector_type(8)));
typedef __attribute__((address_space(3))) float lds_float;

#if defined(__HIP_DEVICE_COMPILE__) && \
    __has_builtin(__builtin_amdgcn_tensor_load_to_lds) && \
    __has_builtin(__builtin_amdgcn_s_wait_tensorcnt)
#define HAVE_TDM 1
#else
#define HAVE_TDM 0
#endif

#define D       64
#define MAXREL  32          // reference: n_rel = 32
#define MIN_NORM 1e-15f

__device__ __forceinline__ float wsum(float v) {
    // wave32 butterfly reduction: every lane ends with the full sum
    v += __shfl_xor(v, 16, 32);
    v += __shfl_xor(v,  8, 32);
    v += __shfl_xor(v,  4, 32);
    v += __shfl_xor(v,  2, 32);
    v += __shfl_xor(v,  1, 32);
    return v;
}
__device__ __forceinline__ float dot2(float2 a, float2 b) {
    return wsum(a.x * b.x + a.y * b.y);
}
__device__ __forceinline__ float tclip(float x) {
    return tanhf(fminf(fmaxf(x, -15.f), 15.f));
}

// ---------------------------------------------------------------------------
__global__ void zero_kernel(float* __restrict__ out, int n_out,
                            float* __restrict__ counts, int n_ent) {
    int i = blockIdx.x * blockDim.x + threadIdx.x;
    if (i < n_out) out[i]    = 0.f;
    if (i < n_ent) counts[i] = 0.f;
}

// ---------------------------------------------------------------------------
__global__ void __launch_bounds__(256)
edge_kernel(const float* __restrict__ ego,
            const int*   __restrict__ heads,
            const int*   __restrict__ tails,
            const int*   __restrict__ etype,
            const float* __restrict__ rel,
            int n_edges, int n_rel,
            float* __restrict__ out_sums,
            float* __restrict__ counts) {
    __shared__ float lds_rel[MAXREL * D];        // 8 KB (single LDS alloc)
    const int nr = n_rel < MAXREL ? n_rel : MAXREL;

#if HAVE_TDM
    if (threadIdx.x == 0) {
        // --- Tensor DMA Descriptor (CDNA5 ISA), 2-D tile ---
        // tile = nr rows x 64 cols of f32, contiguous in global, dense in LDS.
        u32 lds_base = (u32)(u64)(lds_float*)&lds_rel[0];  // LDS byte offset
        u64 ga = (u64)rel;
        u32x4 g0;
        g0.x = 1u;                                   // count=1 (valid user D#)
        g0.y = lds_base;                             // lds_addr
        g0.z = (u32)(ga & 0xFFFFFFFFull);            // global_addr[31:0]
        g0.w = (u32)((ga >> 32) & 0x1FFFFFFull)      // global_addr[56:32]
             | 0x80000000u;                          // type = 2 ("image")
        u32 dim0 = (u32)D;                           // tensor_dim0 (elements)
        u32 dim1 = (u32)nr;                          // tensor_dim1 (rows)
        i32x8 g1;
        g1[0] = (int)(2u << 16);                     // data_size = 2 (4 bytes)
        g1[1] = (int)((dim0 & 0xFFFFu) << 16);       // tensor_dim0[15:0]
        g1[2] = (int)(((dim0 >> 16) & 0xFFFFu) |
                      ((dim1 & 0xFFFFu) << 16));     // dim0[31:16] | dim1[15:0]
        g1[3] = (int)(((dim1 >> 16) & 0xFFFFu) |
                      ((u32)D << 16));               // dim1[31:16] | tile_dim0
        g1[4] = (int)(dim1 & 0xFFFFu);               // tile_dim1 | tile_dim2=0
        g1[5] = (int)D;                              // tensor_dim0_stride lo
        g1[6] = 0;                                   // stride hi | dim1_stride lo
        g1[7] = 0;                                   // dim1_stride hi
        i32x4 gz4 = {0, 0, 0, 0};                    // groups 2/3 unused (2-D)
        i32x8 gz8 = {0, 0, 0, 0, 0, 0, 0, 0};        // extra group (clang-23 form)
        __builtin_amdgcn_tensor_load_to_lds(g0, g1, gz4, gz4, gz8, 0);
        __builtin_amdgcn_s_wait_tensorcnt((short)0); // TDM done -> LDS valid
    }
#else
    for (int i = threadIdx.x; i < nr * D; i += blockDim.x) lds_rel[i] = rel[i];
#endif
    __syncthreads();

    const int lane   = threadIdx.x & 31;
    const int wave   = (blockIdx.x * blockDim.x + threadIdx.x) >> 5;
    const int nwaves = (gridDim.x * blockDim.x) >> 5;
    const float maxn = 1.0f - 1e-3f;                 // project() boundary

    for (int e = wave; e < n_edges; e += nwaves) {
        if (e + nwaves < n_edges) {                  // gfx1250 global_prefetch
            __builtin_prefetch(&heads[e + nwaves], 0, 0);
            __builtin_prefetch(&tails[e + nwaves], 0, 0);
        }
        const int hidx = heads[e];
        const int tidx = tails[e];
        const int ridx = etype[e];

        const float2 h = *(const float2*)(ego + (size_t)hidx * D + 2 * lane);
        const float2 t = *(const float2*)(ego + (size_t)tidx * D + 2 * lane);
        const float2 r = *(const float2*)(lds_rel + ridx * D + 2 * lane);

        // ---- expmap0(h) -> hh ------------------------------------------------
        const float h2  = dot2(h, h);
        const float nh  = fmaxf(sqrtf(h2), MIN_NORM);
        const float sh  = tclip(nh) / nh;
        const float2 hh = make_float2(sh * h.x, sh * h.y);
        const float hh2 = sh * sh * h2;
        const float lam = 2.0f / fmaxf(1.0f - hh2, MIN_NORM);
        const float one_m_hh2 = 1.0f - hh2;

        // ---- expmap(t, hh) -> htv  (mobius_add(hh, tanh(.5*lam*|t|)*t/|t|)) --
        const float t2  = dot2(t, t);
        const float nt  = fmaxf(sqrtf(t2), MIN_NORM);
        const float st  = tclip(0.5f * lam * nt) / nt;
        const float y2t = st * st * t2;
        const float xyt = sh * st * dot2(h, t);
        const float at  = 1.f + 2.f * xyt + y2t;
        const float dent = fmaxf(1.f + 2.f * xyt + hh2 * y2t, MIN_NORM);
        const float2 htv = make_float2((at * hh.x + one_m_hh2 * st * t.x) / dent,
                                       (at * hh.y + one_m_hh2 * st * t.y) / dent);

        // ---- expmap(r, hh) -> hrv -------------------------------------------
        const float r2  = dot2(r, r);
        const float nrm = fmaxf(sqrtf(r2), MIN_NORM);
        const float sr  = tclip(0.5f * lam * nrm) / nrm;
        const float y2r = sr * sr * r2;
        const float xyr = sh * sr * dot2(h, r);
        const float ar  = 1.f + 2.f * xyr + y2r;
        const float denr = fmaxf(1.f + 2.f * xyr + hh2 * y2r, MIN_NORM);
        const float2 hrv = make_float2((ar * hh.x + one_m_hh2 * sr * r.x) / denr,
                                       (ar * hh.y + one_m_hh2 * sr * r.y) / denr);

        // ---- mobius_add(htv, hrv) -> m --------------------------------------
        const float x2 = dot2(htv, htv);
        const float y2 = dot2(hrv, hrv);
        const float xy = dot2(htv, hrv);
        const float denm = fmaxf(1.f + 2.f * xy + x2 * y2, MIN_NORM);
        const float ca = (1.f + 2.f * xy + y2) / denm;
        const float cb = (1.f - x2) / denm;
        const float2 m = make_float2(ca * htv.x + cb * hrv.x,
                                     ca * htv.y + cb * hrv.y);

        // ---- project(m) -> p ------------------------------------------------
        const float m2  = dot2(m, m);
        const float nm  = fmaxf(sqrtf(m2), MIN_NORM);
        const float psc = (nm > maxn) ? (maxn / nm) : 1.0f;
        const float2 p  = make_float2(psc * m.x, psc * m.y);
        const float p2  = psc * psc * m2;

        // ---- logmap(p, hh): sub = mobius_add(-hh, p) ------------------------
        const float xys = -dot2(hh, p);
        const float dens = fmaxf(1.f + 2.f * xys + hh2 * p2, MIN_NORM);
        const float cas = (1.f + 2.f * xys + p2) / dens;
        const float cbs = one_m_hh2 / dens;
        const float2 sub = make_float2(cbs * p.x - cas * hh.x,
                                       cbs * p.y - cas * hh.y);
        const float s2 = dot2(sub, sub);
        const float ns = fmaxf(sqrtf(s2), MIN_NORM);
        const float nc = fminf(fmaxf(ns, -1.f + 1e-7f), 1.f - 1e-7f);
        const float art = 0.5f * logf((1.f + nc) / (1.f - nc));  // artanh
        const float coef = fmaxf(1.f - hh2, MIN_NORM) * art / ns; // (2/lam)*...

        // ---- scatter-add into sums[head], count++ ---------------------------
        float* o = out_sums + (size_t)hidx * D + 2 * lane;
        unsafeAtomicAdd(o,     coef * sub.x);
        unsafeAtomicAdd(o + 1, coef * sub.y);
        if (lane == 0) unsafeAtomicAdd(counts + hidx, 1.0f);
    }
}

// ---------------------------------------------------------------------------
__global__ void finalize_kernel(float* __restrict__ out,
                                const float* __restrict__ counts, int total) {
    int i = blockIdx.x * blockDim.x + threadIdx.x;
    if (i < total) out[i] = out[i] / fmaxf(counts[i >> 6], 1.0f);  // d = 64
}

// ---------------------------------------------------------------------------
extern "C" void kernel_launch(void* const* d_in, const int* in_sizes, int n_in,
                              void* d_out, int out_size, void* d_ws, size_t ws_size,
                              hipStream_t stream) {
    const float* ego  = (const float*)d_in[0];   // [n_ent, 64] f32
    const int*   eidx = (const int*)d_in[1];     // [2, E]
    const int*   etyp = (const int*)d_in[2];     // [E]
    const float* rel  = (const float*)d_in[3];   // [n_rel, 64] f32

    const int n_ent = in_sizes[0] / D;
    const int E     = in_sizes[2];
    const int n_rel = in_sizes[3] / D;

    float* out    = (float*)d_out;               // [n_ent, 64] f32
    float* counts = (float*)d_ws;                // n_ent floats of scratch

    const int zmax = out_size > n_ent ? out_size : n_ent;
    zero_kernel<<<(zmax + 255) / 256, 256, 0, stream>>>(out, out_size,
                                                        counts, n_ent);

    // persistent: 8192 blocks x 8 waves; each wave grid-strides over edges
    edge_kernel<<<8192, 256, 0, stream>>>(ego, eidx, eidx + E, etyp, rel,
                                          E, n_rel, out, counts);

    finalize_kernel<<<(out_size + 255) / 256, 256, 0, stream>>>(out, counts,
                                                                out_size);
}